// HeadDetection_81406810128711
// MI455X (gfx1250) — compile-verified
//
#include <hip/hip_runtime.h>

// CDNA5 / gfx1250: wave32, WMMA 16x16x4 f32 (full-precision matrix pipe).
typedef float v2f __attribute__((ext_vector_type(2)));
typedef float v8f __attribute__((ext_vector_type(8)));

#define IC 256
#define OC 255

// ---------------------------------------------------------------------------
// Repack weights OIHW [255,256,3,3] -> K-major [t][ic][oc(padded 256)] so that
// WMMA A-fragment loads are coalesced (16 consecutive lanes = 64 consecutive
// bytes). Padded oc row 255 is zero.
// ---------------------------------------------------------------------------
__global__ __launch_bounds__(256) void repack_weights_k(
    const float* __restrict__ w, float* __restrict__ wp) {
  int idx = blockIdx.x * 256 + threadIdx.x;   // flat over [9][256][256]
  if (idx >= 9 * IC * 256) return;
  int oc   = idx & 255;
  int rest = idx >> 8;
  int ic   = rest & 255;
  int t    = rest >> 8;
  float v = (oc < OC) ? w[((size_t)oc * IC + ic) * 9 + t] : 0.0f;
  wp[idx] = v;
}

// ---------------------------------------------------------------------------
// Implicit-GEMM 3x3 SAME conv, fp32, via V_WMMA_F32_16X16X4_F32.
//   M = 256 (oc padded), N = B*S*S pixels, K = 9 taps x 256 ic (tap-major).
// Block: 256 threads (8 waves) -> full-M x BN pixel tile of one image row.
// Wave w owns M rows [32w, 32w+32) as 2 WMMA M-tiles x NT N-tiles.
// B (im2col) staged in LDS with halo: [3][BN+2][ic:32] padded stride 34.
// ---------------------------------------------------------------------------
template <int S, int BN, bool PACKED>
__global__ __launch_bounds__(256) void conv3x3_wmma_k(
    const float* __restrict__ in,   // [2,256,S,S]
    const float* __restrict__ w,    // [255,256,3,3]  (native, used if !PACKED)
    const float* __restrict__ wp,   // [9][256][256]  (packed, used if PACKED)
    float* __restrict__ out)        // [2,255,S,S]
{
  constexpr int NT  = BN / 16;   // WMMA N-tiles per wave
  constexpr int ICB = 32;        // input channels staged per pass
  constexpr int TW  = BN + 2;    // tile width incl. halo
  constexpr int ICP = 34;        // padded ic stride (floats): conflict-free b64
  __shared__ float lds[3 * TW * ICP];

  const int tid   = threadIdx.x;
  const int wave  = tid >> 5;
  const int lane  = tid & 31;
  const int lhalf = lane >> 4;   // 0: lanes 0-15, 1: lanes 16-31
  const int l16   = lane & 15;

  constexpr int segPerRow = S / BN;
  int b = blockIdx.x;
  const int x0 = (b % segPerRow) * BN;
  b /= segPerRow;
  const int y = b % S;
  const int n = b / S;

  const float* inN = in + (size_t)n * IC * S * S;

  v8f acc[2][NT];
  {
    v8f z = {};
#pragma unroll
    for (int mt = 0; mt < 2; ++mt)
#pragma unroll
      for (int j = 0; j < NT; ++j) acc[mt][j] = z;
  }

  const int mBase = wave * 32;

  for (int ic0 = 0; ic0 < IC; ic0 += ICB) {
    __syncthreads();   // LDS reuse across ic-blocks
    // ---- stage im2col input tile [3][TW][ICB] (ic-fast, padded) ----
    constexpr int TOT = 3 * TW * ICB;
    for (int e = tid; e < TOT; e += 256) {
      int ic  = e / (3 * TW);
      int rem = e - ic * (3 * TW);
      int r   = rem / TW;
      int c   = rem - r * TW;
      int yy  = y - 1 + r;
      int xx  = x0 - 1 + c;
      float v = 0.0f;
      if ((unsigned)yy < (unsigned)S && (unsigned)xx < (unsigned)S)
        v = inN[(size_t)(ic0 + ic) * (S * S) + yy * S + xx];
      lds[(r * TW + c) * ICP + ic] = v;
    }
    __syncthreads();

    // ---- GEMM over this ic-block: 9 taps x (ICB/4) WMMA K-steps ----
    for (int t = 0; t < 9; ++t) {
      const int ky = t / 3;
      const int kx = t - ky * 3;
#pragma unroll
      for (int ks = 0; ks < ICB / 4; ++ks) {
        const int icK = ks * 4 + 2 * lhalf;   // this lane's K pair base

        // B fragments: 4x16 f32, lane l: N=l%16, K=j+2*(l>=16) -> ds_load_b64
        v2f bf[NT];
#pragma unroll
        for (int j = 0; j < NT; ++j) {
          const int c = j * 16 + l16 + kx;
          bf[j] = *(const v2f*)&lds[(ky * TW + c) * ICP + icK];
        }

        // A fragments: 16x4 f32, lane l: M=l%16, K=j+2*(l>=16)
        v2f af[2];
#pragma unroll
        for (int mt = 0; mt < 2; ++mt) {
          const int m = mBase + mt * 16 + l16;
          if (PACKED) {
            const float* p = wp + ((size_t)(t * IC + (ic0 + icK)) << 8) + m;
            af[mt].x = p[0];      // K = icK   (coalesced across lanes)
            af[mt].y = p[256];    // K = icK+1
          } else {
            const int mw = (m < OC) ? m : OC - 1;   // pad row duplicates 254
            const float* p = w + ((size_t)mw * IC + (ic0 + icK)) * 9 + t;
            af[mt].x = p[0];
            af[mt].y = p[9];
          }
        }

#pragma unroll
        for (int mt = 0; mt < 2; ++mt)
#pragma unroll
          for (int j = 0; j < NT; ++j)
            acc[mt][j] = __builtin_amdgcn_wmma_f32_16x16x4_f32(
                false, af[mt], false, bf[j],
                (short)0, acc[mt][j], false, false);
      }
    }
  }

  // ---- store D: vgpr r, lane l -> M = r + 8*(l>=16), N = l%16 ----
  float* outN = out + (size_t)n * OC * S * S;
#pragma unroll
  for (int mt = 0; mt < 2; ++mt) {
    const int mRowBase = mBase + mt * 16 + 8 * lhalf;
#pragma unroll
    for (int j = 0; j < NT; ++j) {
      const int xc = x0 + j * 16 + l16;
#pragma unroll
      for (int r = 0; r < 8; ++r) {
        const int m = mRowBase + r;
        if (m < OC)
          outN[(size_t)m * (S * S) + y * S + xc] = acc[mt][j][r];
      }
    }
  }
}

// ---------------------------------------------------------------------------
extern "C" void kernel_launch(void* const* d_in, const int* in_sizes, int n_in,
                              void* d_out, int out_size, void* d_ws, size_t ws_size,
                              hipStream_t stream) {
  (void)in_sizes; (void)n_in; (void)out_size;
  // Output order: conv(feat4,W0), conv(feat3,W1), ..., conv(feat0,W4)
  const float* feats[5] = {
      (const float*)d_in[4],   // S=256
      (const float*)d_in[3],   // S=128
      (const float*)d_in[2],   // S=64
      (const float*)d_in[1],   // S=32
      (const float*)d_in[0]    // S=16
  };
  const float* Wn[5] = {
      (const float*)d_in[5], (const float*)d_in[6], (const float*)d_in[7],
      (const float*)d_in[8], (const float*)d_in[9]
  };
  float* out = (float*)d_out;

  const size_t WPACK = (size_t)9 * IC * 256;                 // floats per level
  const bool packed  = ws_size >= 5 * WPACK * sizeof(float); // 11.8 MB needed
  float* wp = (float*)d_ws;

  if (packed) {
    const int rblocks = (9 * IC * 256 + 255) / 256;
    for (int L = 0; L < 5; ++L)
      repack_weights_k<<<rblocks, 256, 0, stream>>>(Wn[L], wp + (size_t)L * WPACK);
  }

  size_t off = 0;
#define LAUNCH(Sv, BNv, L)                                                    \
  do {                                                                        \
    dim3 g(2 * Sv * (Sv / BNv));                                              \
    if (packed)                                                               \
      conv3x3_wmma_k<Sv, BNv, true><<<g, 256, 0, stream>>>(                   \
          feats[L], Wn[L], wp + (size_t)L * WPACK, out + off);                \
    else                                                                      \
      conv3x3_wmma_k<Sv, BNv, false><<<g, 256, 0, stream>>>(                  \
          feats[L], Wn[L], (const float*)nullptr, out + off);                 \
    off += (size_t)2 * OC * Sv * Sv;                                          \
  } while (0)

  LAUNCH(256, 64, 0);
  LAUNCH(128, 64, 1);
  LAUNCH(64,  64, 2);
  LAUNCH(32,  32, 3);
  LAUNCH(16,  16, 4);
#undef LAUNCH
}